// Symmetrizer_61117384622598
// MI455X (gfx1250) — compile-verified
//
#include <hip/hip_runtime.h>
#include <hip/hip_bf16.h>

typedef unsigned int u32;
typedef unsigned long long u64;
typedef __attribute__((ext_vector_type(4))) u32 v4u;
typedef __attribute__((ext_vector_type(8))) int v8i;
typedef __attribute__((ext_vector_type(4))) int v4i;

#define TILE_DW      160                      // 20 angular * 8 channels (dwords)
#define TILE_DW_PAD  180                      // +4 dwords pad every 32 dwords
#define CHUNK_TILES  32
#define CHUNK_DW     (CHUNK_TILES * TILE_DW)      // 5120 dwords / chunk
#define CHUNK_DW_PAD (CHUNK_TILES * TILE_DW_PAD)  // 5760 dwords / chunk in LDS
#define CHUNK_BYTES_PAD (CHUNK_DW_PAD * 4)        // 23040 B

#if __has_builtin(__builtin_amdgcn_tensor_load_to_lds)
#define USE_TDM 1
#else
#define USE_TDM 0
#endif

__device__ __forceinline__ void sym_wait_tensorcnt1() {
#if __has_builtin(__builtin_amdgcn_s_wait_tensorcnt)
  __builtin_amdgcn_s_wait_tensorcnt(1);
#else
  asm volatile("s_wait_tensorcnt 0x1" ::: "memory");
#endif
}
__device__ __forceinline__ void sym_wait_tensorcnt0() {
#if __has_builtin(__builtin_amdgcn_s_wait_tensorcnt)
  __builtin_amdgcn_s_wait_tensorcnt(0);
#else
  asm volatile("s_wait_tensorcnt 0x0" ::: "memory");
#endif
}

#if USE_TDM
// Issue one TDM 2D tile load: rows x 160 dwords, contiguous in memory,
// written to LDS with 4 dwords of padding after every 32 dwords
// (pad_interval code 4 = 32 dwords, pad_amount code 3 = 4 dwords).
__device__ __forceinline__ void tdm_load_chunk(const float* gsrc, u32 lds_byte_off, int rows) {
  u64 ga = (u64)(const void*)gsrc;
  v4u g0;
  g0[0] = 1u;                                            // count=1 (valid), user mode, no gather
  g0[1] = lds_byte_off;                                  // lds_addr [63:32]
  g0[2] = (u32)ga;                                       // global_addr [95:64]
  g0[3] = ((u32)(ga >> 32) & 0x1FFFFFFu) | 0x80000000u;  // addr[56:32] | type=2
  v8i g1;
  g1[0] = (2 << 16)    // data_size = 4 bytes
        | (1 << 20)    // pad_enable
        | (4 << 22)    // pad_interval: 32 dwords
        | (3 << 25);   // pad_amount: 4 dwords
  g1[1] = (TILE_DW & 0xFFFF) << 16;                      // tensor_dim0[15:0]
  g1[2] = (TILE_DW >> 16) | ((rows & 0xFFFF) << 16);     // dim0 hi | tensor_dim1[15:0] (OOB rows -> 0)
  g1[3] = (TILE_DW << 16);                               // tensor_dim1 hi=0 | tile_dim0=160
  g1[4] = CHUNK_TILES;                                   // tile_dim1=32, tile_dim2=0
  g1[5] = TILE_DW;                                       // tensor_dim0_stride lo32 (rows contiguous)
  g1[6] = 0;
  g1[7] = 0;
  v4i gz4 = {0, 0, 0, 0};                                // groups 2/3: dims/strides 0, no iterate
  v8i gz8 = {0, 0, 0, 0, 0, 0, 0, 0};                    // trailing group (unused, zero)
  __builtin_amdgcn_tensor_load_to_lds(g0, g1, gz4, gz4, gz8, 0);
}
#endif

// LDS padded address of (tile lt, angular t, channel c):
//   lt*180 + 8*t + 4*(t>>2) + c      (conflict-free across the 4 tiles of a wave)
__device__ __forceinline__ void compute_store_tile(const float* __restrict__ sb,
                                                   float* __restrict__ out,
                                                   int gtile, int lt, int c) {
  const float* b = sb + lt * TILE_DW_PAD + c;
#define AT(t) b[8 * (t) + 4 * ((t) >> 2)]
  const float a0  = AT(0);
  const float a1  = AT(1),  a2  = AT(2),  a3  = AT(3);
  const float a4  = AT(4),  a5  = AT(5),  a6  = AT(6);
  const float a7  = AT(7),  a8  = AT(8),  a9  = AT(9);
  const float a10 = AT(10), a11 = AT(11), a12 = AT(12), a13 = AT(13), a14 = AT(14);
  const float a15 = AT(15), a16 = AT(16), a17 = AT(17), a18 = AT(18), a19 = AT(19);
#undef AT

  // nu=2 contractions (multinomial-weighted sums of squares)
  const float col1 = a1 * a1 + a2 * a2 + a3 * a3;
  const float col2 = a4 * a4 + a7 * a7 + a9 * a9 + 2.f * (a5 * a5 + a6 * a6 + a8 * a8);
  const float col3 = a10 * a10 + a16 * a16 + a19 * a19
                   + 3.f * (a11 * a11 + a12 * a12 + a13 * a13 + a15 * a15 + a17 * a17 + a18 * a18)
                   + 6.f * (a14 * a14);

  // Q = deg-2 block as symmetric 3x3, r = deg-1 block
  auto quadQ = [&](float px, float py, float pz) {
    return a4 * px * px + a7 * py * py + a9 * pz * pz
         + 2.f * (a5 * px * py + a6 * px * pz + a8 * py * pz);
  };
  auto dotR = [&](float px, float py, float pz) { return a1 * px + a2 * py + a3 * pz; };

  const float col4 = quadQ(a1, a2, a3);                     // (0,1,1): r^T Q r
  const float col5 =       a4 * dotR(a10, a11, a12)         // (0,1,2)
                   + 2.f * a5 * dotR(a11, a13, a14)
                   + 2.f * a6 * dotR(a12, a14, a15)
                   +       a7 * dotR(a13, a16, a17)
                   + 2.f * a8 * dotR(a14, a17, a18)
                   +       a9 * dotR(a15, a18, a19);
  const float s55 = a5 * a5, s66 = a6 * a6, s88 = a8 * a8;  // (1,1,1): tr(Q^3)
  const float col6 = a4 * a4 * a4 + a7 * a7 * a7 + a9 * a9 * a9
                   + 3.f * (a4 * (s55 + s66) + a7 * (s55 + s88) + a9 * (s66 + s88))
                   + 6.f * (a5 * a6 * a8);
  const float col7 =       quadQ(a10, a11, a12)             // (1,1,2)
                   + 2.f * quadQ(a11, a13, a14)
                   + 2.f * quadQ(a12, a14, a15)
                   +       quadQ(a13, a16, a17)
                   + 2.f * quadQ(a14, a17, a18)
                   +       quadQ(a15, a18, a19);

  float* ob = out + (size_t)gtile * 64 + c;
  ob[0]  = a0;   ob[8]  = col1; ob[16] = col2; ob[24] = col3;
  ob[32] = col4; ob[40] = col5; ob[48] = col6; ob[56] = col7;
}

__global__ __launch_bounds__(256) void Symmetrizer_cdna5_kernel(const float* __restrict__ in,
                                                                float* __restrict__ out,
                                                                int numTiles) {
#if USE_TDM
  __shared__ float sbuf[2 * CHUNK_DW_PAD];   // double-buffered, 46080 B
#else
  __shared__ float sbuf[CHUNK_DW_PAD];
#endif
  const int tid = threadIdx.x;
  const int lt = tid >> 3;   // local tile 0..31
  const int c = tid & 7;     // channel 0..7
  const int numChunks = (numTiles + CHUNK_TILES - 1) / CHUNK_TILES;

#if USE_TDM
  const u32 ldsBase = (u32)(u64)(const void*)&sbuf[0];  // flat low 32 bits = LDS byte offset
  const int q0 = blockIdx.x;
  if (tid < 32 && q0 < numChunks) {   // wave 0 only: TDM ignores EXEC, must be branched over
    int rows = numTiles - q0 * CHUNK_TILES;
    rows = rows > CHUNK_TILES ? CHUNK_TILES : rows;
    tdm_load_chunk(in + (size_t)q0 * CHUNK_DW, ldsBase, rows);
  }
  int parity = 0;
  for (int q = q0; q < numChunks; q += gridDim.x, parity ^= 1) {
    const int nq = q + (int)gridDim.x;
    if (tid < 32) {
      if (nq < numChunks) {           // prefetch next chunk into other buffer
        int rows = numTiles - nq * CHUNK_TILES;
        rows = rows > CHUNK_TILES ? CHUNK_TILES : rows;
        tdm_load_chunk(in + (size_t)nq * CHUNK_DW,
                       ldsBase + (u32)((parity ^ 1) * CHUNK_BYTES_PAD), rows);
        sym_wait_tensorcnt1();        // current chunk (issued earlier, in-order) complete
      } else {
        sym_wait_tensorcnt0();
      }
    }
    __syncthreads();                  // LDS data visible to all 8 waves
    const int gtile = q * CHUNK_TILES + lt;
    if (gtile < numTiles) compute_store_tile(sbuf + parity * CHUNK_DW_PAD, out, gtile, lt, c);
    __syncthreads();                  // buffer free before wave 0 refills it next round
  }
#else
  for (int q = blockIdx.x; q < numChunks; q += gridDim.x) {
    int rows = numTiles - q * CHUNK_TILES;
    rows = rows > CHUNK_TILES ? CHUNK_TILES : rows;
    const int maxv = rows * (TILE_DW / 4);
    const float4* gsrc = (const float4*)(in + (size_t)q * CHUNK_DW);
    for (int i = tid; i < CHUNK_DW / 4; i += 256) {
      float4 v;
      if (i < maxv) v = gsrc[i];
      else { v.x = 0.f; v.y = 0.f; v.z = 0.f; v.w = 0.f; }
      const int w = i * 4;
      *(float4*)&sbuf[w + 4 * (w >> 5)] = v;   // same padded layout, stays 16B aligned
    }
    __syncthreads();
    const int gtile = q * CHUNK_TILES + lt;
    if (gtile < numTiles) compute_store_tile(sbuf, out, gtile, lt, c);
    __syncthreads();
  }
#endif
}

extern "C" void kernel_launch(void* const* d_in, const int* in_sizes, int n_in,
                              void* d_out, int out_size, void* d_ws, size_t ws_size,
                              hipStream_t stream) {
  (void)n_in; (void)out_size; (void)d_ws; (void)ws_size;
  const float* in = (const float*)d_in[0];
  float* out = (float*)d_out;
  const long long numTiles = (long long)in_sizes[0] / TILE_DW;   // N * N_RADIAL = 320000
  const int numChunks = (int)((numTiles + CHUNK_TILES - 1) / CHUNK_TILES);
  const int blocks = numChunks < 2560 ? numChunks : 2560;        // persistent, multi-chunk pipeline
  hipLaunchKernelGGL(Symmetrizer_cdna5_kernel, dim3(blocks), dim3(256), 0, stream,
                     in, out, (int)numTiles);
}